// RNN_43920335569315
// MI455X (gfx1250) — compile-verified
//
#include <hip/hip_runtime.h>

// Fused tanh-RNN for MI455X (gfx1250, wave32).
// One wave owns 16 batch chains (WMMA N-columns); recurrence + input proj +
// bias all run inside v_wmma_f32_16x16x4_f32 accumulator chains. The x-part
// WMMAs are software-pipelined one step ahead so only the two W_hh WMMAs +
// tanh + half-swap sit on the serial critical path. Memory-bound problem
// (~134 MB total): x streamed once, out staged in LDS for coalesced b128
// stores. 512 single-wave workgroups = max WGP coverage for 8192 chains.

typedef __attribute__((ext_vector_type(2))) float v2f;
typedef __attribute__((ext_vector_type(8))) float v8f;

#define RNN_B 8192
#define RNN_T 512
#define RNN_H 7
#define TILE_T 16
#define N_TILES (RNN_T / TILE_T)

static __device__ __forceinline__ float fast_tanh(float x) {
#if __has_builtin(__builtin_amdgcn_tanhf)
  return __builtin_amdgcn_tanhf(x);  // v_tanh_f32 (TRANS, co-executes with XDL)
#elif __has_builtin(__builtin_amdgcn_tanh_f32)
  return __builtin_amdgcn_tanh_f32(x);
#else
  float y;
  asm volatile("v_tanh_f32 %0, %1" : "=v"(y) : "v"(x));
  return y;
#endif
}

static __device__ __forceinline__ v8f wmma4(v2f a, v2f b, v8f c) {
  // (neg_a, A, neg_b, B, c_mod, C, reuse_a, reuse_b)
  return __builtin_amdgcn_wmma_f32_16x16x4_f32(false, a, false, b, (short)0, c,
                                               false, false);
}

// Broadcast lane n's value (n = lane&15) of the LOW half to lane 16+n of the
// HIGH half. permlanex16 with identity selects swaps the 16-lane halves as a
// pure VALU op (no LDS latency); bpermute fallback otherwise.
static __device__ __forceinline__ float bcast_lo16(float v, int pidx) {
#if __has_builtin(__builtin_amdgcn_permlanex16)
  (void)pidx;
  return __int_as_float(__builtin_amdgcn_permlanex16(
      __float_as_int(v), __float_as_int(v), 0x76543210, 0xfedcba98, false,
      false));
#else
  return __int_as_float(__builtin_amdgcn_ds_bpermute(pidx, __float_as_int(v)));
#endif
}

__global__ __launch_bounds__(32) void rnn_fused_wmma(
    const float* __restrict__ x,   // [B,T,7]
    const float* __restrict__ Wih, // [7,7]
    const float* __restrict__ Whh, // [7,7]
    const float* __restrict__ bih, // [7]
    const float* __restrict__ bhh, // [7]
    const float* __restrict__ Wfc, // [1,7]
    const float* __restrict__ bfc, // [1]
    float* __restrict__ out,       // [B,T]
    float* __restrict__ hid)       // [B,7]
{
  // 32 rows (one per lane; rows 16-31 are dump rows so the per-step store is
  // unconditional -> no exec juggling). Row stride 20: 16B-aligned float4
  // reads for the flush, conflict-light writes.
  __shared__ float obuf[32][20];

  const int lane = threadIdx.x & 31;
  const int n    = lane & 15;  // N column (batch within the wave's group)
  const int s    = lane >> 4;  // lane-half select
  const bool lo  = (s == 0);
  const int b0   = blockIdx.x * 16;

  // ---- A operands: 16x4 f32 A layout (ISA 7.12.2):
  // lanes0-15: M=lane, v0=K0, v1=K1 ; lanes16-31: M=lane-16, v0=K2, v1=K3.
  auto aelem = [&](const float* W, int m, int k) -> float {
    return (m < RNN_H && k < RNN_H) ? W[m * RNN_H + k] : 0.0f;
  };
  const int k0 = 2 * s;
  v2f a_ih_lo, a_ih_hi, a_hh_lo, a_hh_hi;
  a_ih_lo.x = aelem(Wih, n, k0);     a_ih_lo.y = aelem(Wih, n, k0 + 1);
  a_ih_hi.x = aelem(Wih, n, k0 + 4); a_ih_hi.y = aelem(Wih, n, k0 + 5);
  a_hh_lo.x = aelem(Whh, n, k0);     a_hh_lo.y = aelem(Whh, n, k0 + 1);
  a_hh_hi.x = aelem(Whh, n, k0 + 4); a_hh_hi.y = aelem(Whh, n, k0 + 5);

  // ---- C bias in D layout (row r lanes0-15 / row r+8 lanes16-31).
  v8f cb;
#pragma unroll
  for (int r = 0; r < 8; ++r) {
    const float bv = (r < RNN_H) ? (bih[r] + bhh[r]) : 0.0f;  // uniform
    cb[r] = lo ? bv : 0.0f;
  }

  // ---- Per-lane x element offsets for the B_x layout:
  // B vgpr j holds rows K=j (lanes0-15) and K=j+2 (lanes16-31).
  const float* xt = x + (size_t)(b0 + n) * RNN_T * RNN_H;
  const int i0 = k0;      // B_lo.x : i = 0 / 2
  const int i1 = k0 + 1;  // B_lo.y : i = 1 / 3
  const int i2 = k0 + 4;  // B_hi.x : i = 4 / 6
  // B_hi.y : i = 5 / (7 -> 0): load i=5 both halves (in-bounds), zero upper.

  float wf[RNN_H];
#pragma unroll
  for (int r = 0; r < RNN_H; ++r) wf[r] = Wfc[r];
  const float ob0 = bfc[0];
  const int pidx = n << 2;  // bpermute fallback byte index

  auto load_bx = [&](const float* p, v2f& blo, v2f& bhi) {
    blo.x = p[i0];
    blo.y = p[i1];
    bhi.x = p[i2];
    const float x5 = p[5];
    bhi.y = lo ? x5 : 0.0f;
  };

  v2f bh_lo = {0.0f, 0.0f};
  v2f bh_hi = {0.0f, 0.0f};
  float h[RNN_H];
#pragma unroll
  for (int r = 0; r < RNN_H; ++r) h[r] = 0.0f;

  // ---- Prologue: x-part accumulator for t=0.
  v2f bx_lo, bx_hi;
  load_bx(xt, bx_lo, bx_hi);
  v8f xacc = wmma4(a_ih_hi, bx_hi, wmma4(a_ih_lo, bx_lo, cb));

  for (int tile = 0; tile < N_TILES; ++tile) {
#pragma unroll
    for (int q = 0; q < TILE_T; ++q) {
      // Issue next step's x loads early (immediate offsets within the tile).
      v2f nx_lo, nx_hi;
      if (q < TILE_T - 1) load_bx(xt + (q + 1) * RNN_H, nx_lo, nx_hi);

      // Serial critical path: just the two W_hh WMMAs.
      v8f acc = wmma4(a_hh_hi, bh_hi, wmma4(a_hh_lo, bh_lo, xacc));

      // x-part for the next step -- independent of acc, overlaps with tanh.
      if (q < TILE_T - 1)
        xacc = wmma4(a_ih_hi, nx_hi, wmma4(a_ih_lo, nx_lo, cb));

#pragma unroll
      for (int r = 0; r < RNN_H; ++r) h[r] = fast_tanh(acc[r]);

      // out[b,t] on lanes 0-15; lanes 16-31 write their private dump row.
      float o = ob0;
#pragma unroll
      for (int r = 0; r < RNN_H; ++r) o = fmaf(h[r], wf[r], o);
      obuf[lane][q] = o;

      // Rebuild B_h: upper half needs rows 2/3/6 broadcast from lanes 0-15.
      const float u2 = bcast_lo16(h[2], pidx);
      const float u3 = bcast_lo16(h[3], pidx);
      const float u6 = bcast_lo16(h[6], pidx);
      bh_lo.x = lo ? h[0] : u2;
      bh_lo.y = lo ? h[1] : u3;
      bh_hi.x = lo ? h[4] : u6;
      bh_hi.y = lo ? h[5] : 0.0f;  // K=7 pad row
    }

    // ---- Coalesced flush: 16 steps x 16 batches as b128 stores.
    __syncthreads();
    const int tbase = tile * TILE_T;
#pragma unroll
    for (int it = 0; it < 2; ++it) {
      const int idx = lane + 32 * it;
      const int nb  = idx >> 2;
      const int tq  = (idx & 3) << 2;
      const float4 v = *(const float4*)&obuf[nb][tq];
      *(float4*)(out + (size_t)(b0 + nb) * RNN_T + tbase + tq) = v;
    }
    __syncthreads();

    xt += TILE_T * RNN_H;
    if (tile < N_TILES - 1) {
      // x-part accumulator for the first step of the next tile.
      load_bx(xt, bx_lo, bx_hi);
      xacc = wmma4(a_ih_hi, bx_hi, wmma4(a_ih_lo, bx_lo, cb));
    }
  }

  // ---- Final hidden state h_T (rows live in lanes 0-15).
  if (lo) {
#pragma unroll
    for (int r = 0; r < RNN_H; ++r)
      hid[(size_t)(b0 + n) * RNN_H + r] = h[r];
  }
}

extern "C" void kernel_launch(void* const* d_in, const int* in_sizes, int n_in,
                              void* d_out, int out_size, void* d_ws,
                              size_t ws_size, hipStream_t stream) {
  (void)in_sizes; (void)n_in; (void)out_size; (void)d_ws; (void)ws_size;
  const float* x   = (const float*)d_in[0];
  const float* Wih = (const float*)d_in[1];
  const float* Whh = (const float*)d_in[2];
  const float* bih = (const float*)d_in[3];
  const float* bhh = (const float*)d_in[4];
  const float* Wfc = (const float*)d_in[5];
  const float* bfc = (const float*)d_in[6];
  float* out = (float*)d_out;                // [B,T]
  float* hid = out + (size_t)RNN_B * RNN_T;  // [1,B,H] appended

  dim3 grid(RNN_B / 16);  // 512 independent 16-chain waves
  dim3 block(32);         // one wave per workgroup
  rnn_fused_wmma<<<grid, block, 0, stream>>>(x, Wih, Whh, bih, bhh, Wfc, bfc,
                                             out, hid);
}